// CoopLatentTreeRL_45681272160770
// MI455X (gfx1250) — compile-verified
//
#include <hip/hip_runtime.h>

typedef __attribute__((ext_vector_type(2))) float v2f_t;
typedef __attribute__((ext_vector_type(8))) float v8f_t;
typedef __attribute__((ext_vector_type(4))) unsigned int u32x4_t;
typedef __attribute__((ext_vector_type(8))) int i32x8_t;
typedef __attribute__((ext_vector_type(4))) int i32x4_t;

#define NB 32
#define NT 48
#define ND 256
#define NH 256

__device__ __forceinline__ float dot4f(float4 a, float4 b) {
  return a.x * b.x + a.y * b.y + a.z * b.z + a.w * b.w;
}
__device__ __forceinline__ float wave_sum32(float v) {
#pragma unroll
  for (int off = 16; off > 0; off >>= 1) v += __shfl_xor(v, off, 32);
  return v;
}
__device__ __forceinline__ float sigf(float x) { return 1.0f / (1.0f + expf(-x)); }

__global__ __launch_bounds__(256, 1) void coop_latent_tree_kernel(
    const float* __restrict__ x, const int* __restrict__ length,
    const float* __restrict__ Ww, const float* __restrict__ bw,
    const float* __restrict__ Wc, const float* __restrict__ bc,
    const float* __restrict__ q, float* __restrict__ out) {
  // Row pools: rows 0..47 = h/c sequence slots, rows 48..94 = hcomp/ccomp slots.
  __shared__ __align__(16) float xs[48][256];   // TDM-staged x tile for this batch row
  __shared__ __align__(16) float hpool[95][256];
  __shared__ __align__(16) float cpool[95][256];
  __shared__ __align__(16) float vbuf[2][1280];
  __shared__ __align__(16) float sh2h[2][256];
  __shared__ __align__(16) float sh2c[2][256];
  __shared__ __align__(16) float qs[256];
  __shared__ float wlog[48];
  __shared__ int posH[48];
  __shared__ int posComp[48];
  __shared__ int allIdx[46];
  __shared__ int ctrl[10];  // 0:idx 2,3:hl rows 4,5:hr rows 6:keep0 7:keep1 8:tgt0 9:tgt1
  __shared__ float scal[2]; // logp_sum, ent_sum

  const int b = blockIdx.x;
  const int tid = threadIdx.x;
  const int wave = tid >> 5;
  const int lane = tid & 31;
  const int khalf = lane >> 4;
  const int nl = lane & 15;

  qs[tid] = q[tid];
  if (tid == 0) {
    for (int s = 0; s < 48; ++s) posH[s] = s;
    for (int s = 0; s < 47; ++s) posComp[s] = 48 + s;
    scal[0] = 0.0f;
    scal[1] = 0.0f;
  }

  // ---------- Stage x[b] (48x256 f32 = 48KB, contiguous) into LDS via the TDM.
#if __has_builtin(__builtin_amdgcn_tensor_load_to_lds)
  if (wave == 0) {
    const unsigned long long gaddr = (unsigned long long)(const void*)(x + (size_t)b * NT * ND);
    const unsigned int lds_addr = (unsigned int)(unsigned long long)(uintptr_t)(&xs[0][0]);
    const unsigned int nelem = NT * ND;  // 12288 4-byte elements, 1-D tile
    // D# group 0: [1:0]=count=1, [63:32]=lds_addr, [120:64]=global_addr, [127:126]=type=2
    u32x4_t g0;
    g0.x = 1u;
    g0.y = lds_addr;
    g0.z = (unsigned int)gaddr;
    g0.w = ((unsigned int)(gaddr >> 32) & 0x01FFFFFFu) | 0x80000000u;
    // D# group 1: [17:16]=data_size=2 (4B); tensor_dim0=12288; tensor_dim1=1;
    //             tile_dim0=12288; tile_dim1/2=0 (unused); tensor_dim0_stride=12288
    i32x8_t g1;
    g1[0] = (int)(2u << 16);
    g1[1] = (int)((nelem & 0xFFFFu) << 16);          // tensor_dim0[15:0] at bits[63:48]
    g1[2] = (int)((nelem >> 16) | (1u << 16));       // tensor_dim0[31:16] | tensor_dim1[15:0]
    g1[3] = (int)((nelem & 0xFFFFu) << 16);          // tile_dim0 at bits[127:112]
    g1[4] = 0;                                       // tile_dim1=0, tile_dim2=0
    g1[5] = (int)nelem;                              // tensor_dim0_stride[31:0]
    g1[6] = 0;
    g1[7] = 0;
    i32x4_t g2 = {0, 0, 0, 0};
    i32x4_t g3 = {0, 0, 0, 0};
    i32x8_t g4 = {0, 0, 0, 0, 0, 0, 0, 0};
    __builtin_amdgcn_tensor_load_to_lds(g0, g1, g2, g3, g4, 0);
    __builtin_amdgcn_s_wait_tensorcnt((short)0);
  }
#else
  for (int e = tid; e < NT * ND; e += 256) {
    ((float*)&xs[0][0])[e] = x[(size_t)b * NT * ND + e];
  }
#endif
  __syncthreads();

  // ---------- GEMM1: state = x[b] @ Ww^T + bw  (48x256 * 256x512) via f32 WMMA
  for (int task = wave; task < 96; task += 8) {
    const int mtile = task >> 5;  // 0..2
    const int ntile = task & 31;  // 0..31
    const float* arow = &xs[mtile * 16 + nl][0];
    const float* brow = Ww + (size_t)(ntile * 16 + nl) * ND;
    v8f_t acc = {};
    for (int ks = 0; ks < 64; ++ks) {
      const int kk = ks * 4 + 2 * khalf;
      v2f_t av = *(const v2f_t*)(arow + kk);
      v2f_t bv = *(const v2f_t*)(brow + kk);
      acc = __builtin_amdgcn_wmma_f32_16x16x4_f32(false, av, false, bv, (short)0, acc, false, false);
    }
    const int col = ntile * 16 + nl;
    const float bias = bw[col];
#pragma unroll
    for (int r = 0; r < 8; ++r) {
      const int m = mtile * 16 + r + 8 * khalf;
      const float val = acc[r] + bias;
      if (ntile < 16) hpool[m][col] = val;
      else            cpool[m][col - 256] = val;
    }
  }
  __syncthreads();

  // ---------- GEMM2: initial candidates: v = [h_t | h_{t+1}] @ Wc^T + bc, then gates
  for (int task = wave; task < 48; task += 8) {
    const int mtile = task >> 4;  // 0..2
    const int dtile = task & 15;  // 0..15
    const int m = mtile * 16 + nl;
    const int m1 = (m + 1 < 48) ? m + 1 : 47;
    v8f_t accv[5];
    {
      v8f_t z = {};
#pragma unroll
      for (int g = 0; g < 5; ++g) accv[g] = z;
    }
    for (int ks = 0; ks < 128; ++ks) {
      const int kk = ks * 4 + 2 * khalf;
      v2f_t av;
      if (kk < 256) av = *(const v2f_t*)(&hpool[m][kk]);
      else          av = *(const v2f_t*)(&hpool[m1][kk - 256]);
#pragma unroll
      for (int g = 0; g < 5; ++g) {
        const int bn = g * 256 + dtile * 16 + nl;
        v2f_t bv = *(const v2f_t*)(Wc + (size_t)bn * 512 + kk);
        accv[g] = __builtin_amdgcn_wmma_f32_16x16x4_f32(false, av, false, bv, (short)0, accv[g], false, false);
      }
    }
#pragma unroll
    for (int r = 0; r < 8; ++r) {
      const int mr = mtile * 16 + r + 8 * khalf;
      if (mr < 47) {
        const int d = dtile * 16 + nl;
        const float gi  = accv[0][r] + bc[d];
        const float gfl = accv[1][r] + bc[256 + d];
        const float gfr = accv[2][r] + bc[512 + d];
        const float gu  = accv[3][r] + bc[768 + d];
        const float go  = accv[4][r] + bc[1024 + d];
        const float cl = cpool[mr][d];
        const float cr = cpool[mr + 1][d];
        const float c2 = cl * sigf(gfl + 1.0f) + cr * sigf(gfr + 1.0f) + tanhf(gu) * sigf(gi);
        hpool[48 + mr][d] = sigf(go) * tanhf(c2);
        cpool[48 + mr][d] = c2;
      }
    }
  }
  __syncthreads();

  const int mylen = length[b];

  // ---------- 46 sequential merge steps
  for (int i = 0; i < 46; ++i) {
    const int S = 47 - i;  // current #candidates

    // (a) logits: w[s] = dot(hcomp[s], q) / 16
    for (int s = wave; s < S; s += 8) {
      const float* hr = &hpool[posComp[s]][0];
      const int kk = lane * 8;
      float4 a0 = *(const float4*)(hr + kk);
      float4 a1 = *(const float4*)(hr + kk + 4);
      float4 q0 = *(const float4*)(qs + kk);
      float4 q1 = *(const float4*)(qs + kk + 4);
      float pq = wave_sum32(dot4f(a0, q0) + dot4f(a1, q1));
      if (lane == 0) wlog[s] = pq * 0.0625f;
    }
    __syncthreads();

    // (b) softmax/mask/argmax/logp/entropy + (c) pointer merges (wave 0)
    if (tid < 32) {
      const float NEGHUGE = -3.0e38f;
      float v0 = (lane < S) ? wlog[lane] : NEGHUGE;
      float v1 = (lane + 32 < S) ? wlog[lane + 32] : NEGHUGE;
      float mx = fmaxf(v0, v1);
#pragma unroll
      for (int off = 16; off > 0; off >>= 1) mx = fmaxf(mx, __shfl_xor(mx, off, 32));
      float e0 = (lane < S) ? expf(v0 - mx) : 0.0f;
      float e1 = (lane + 32 < S) ? expf(v1 - mx) : 0.0f;
      float Z = wave_sum32(e0 + e1);
      float p0 = 0.0f, p1 = 0.0f;
      if (lane < S)      p0 = (e0 / Z) * (((i + 1 + lane) < mylen) ? 1.0f : 0.0f) + 1.0e-20f;
      if (lane + 32 < S) p1 = (e1 / Z) * (((i + 33 + lane) < mylen) ? 1.0f : 0.0f) + 1.0e-20f;
      float Z2 = wave_sum32(p0 + p1);
      p0 /= Z2;
      p1 /= Z2;
      float ec = 0.0f;
      if (lane < S) ec += p0 * logf(p0);
      if (lane + 32 < S) ec += p1 * logf(p1);
      ec = wave_sum32(ec);
      float bv = -1.0f;
      int bi = 0x7fffffff;
      if (lane < S) { bv = p0; bi = lane; }
      if ((lane + 32 < S) && (p1 > bv)) { bv = p1; bi = lane + 32; }
#pragma unroll
      for (int off = 16; off > 0; off >>= 1) {
        float ov = __shfl_xor(bv, off, 32);
        int oi = __shfl_xor(bi, off, 32);
        if (ov > bv || (ov == bv && oi < bi)) { bv = ov; bi = oi; }
      }
      if (lane == 0) {
        scal[0] += logf(bv);
        scal[1] -= ec;
        allIdx[i] = bi;
        ctrl[0] = bi;
        const int idx = bi;
        const int dm = ((i + 1) < mylen) ? 1 : 0;
        const int ShOld = 48 - i;
        if (dm) {  // merge: replace slot idx with candidate row, left-shift the rest
          posH[idx] = posComp[idx];
          for (int s2 = idx + 1; s2 < ShOld - 1; ++s2) posH[s2] = posH[s2 + 1];
        }  // dm==0: implicit truncation (drop last)
        const int Sh = ShOld - 1;  // new h length = 47-i
        for (int s2 = idx; s2 < S - 1; ++s2) posComp[s2] = posComp[s2 + 1];
        const int l0 = (idx > 0) ? idx - 1 : 0;
        const int r1 = (idx + 1 < Sh) ? idx + 1 : Sh - 1;
        ctrl[2] = posH[l0];
        ctrl[3] = posH[idx];
        ctrl[4] = posH[idx];
        ctrl[5] = posH[r1];
        ctrl[6] = (idx == 0) ? 1 : 0;       // keep0
        ctrl[7] = (idx > Sh - 2) ? 1 : 0;   // keep1
        const int i1 = (idx < Sh - 2) ? idx : Sh - 2;
        ctrl[8] = posComp[l0];  // i0 == clip(idx-1,0,Sh-2) == l0
        ctrl[9] = posComp[i1];
      }
    }
    __syncthreads();

    // (d) sh2/sc2: 2-row GEMV v = [hl|hr] @ Wc^T + bc  (wave per output row, coalesced)
    for (int jl = wave; jl < 2560; jl += 8) {
      const int r = (jl >= 1280) ? 1 : 0;
      const int j = jl - r * 1280;
      const int hl = ctrl[2 + r];
      const int hrw = ctrl[4 + r];
      const float* wrow = Wc + (size_t)j * 512;
      __builtin_prefetch(wrow + 8 * 512, 0, 0);
      const int kk = lane * 4;
      float4 w0 = *(const float4*)(wrow + kk);
      float4 w1 = *(const float4*)(wrow + kk + 128);
      float4 w2 = *(const float4*)(wrow + kk + 256);
      float4 w3 = *(const float4*)(wrow + kk + 384);
      float4 a0 = *(const float4*)(&hpool[hl][kk]);
      float4 a1 = *(const float4*)(&hpool[hl][kk + 128]);
      float4 a2 = *(const float4*)(&hpool[hrw][kk]);
      float4 a3 = *(const float4*)(&hpool[hrw][kk + 128]);
      float acc = dot4f(w0, a0) + dot4f(w1, a1) + dot4f(w2, a2) + dot4f(w3, a3);
      acc = wave_sum32(acc);
      if (lane == 0) vbuf[r][j] = acc + bc[j];
    }
    __syncthreads();

    // gates -> sh2h/sh2c
    {
      const int d = tid;
#pragma unroll
      for (int r = 0; r < 2; ++r) {
        const float gi  = vbuf[r][d];
        const float gfl = vbuf[r][256 + d];
        const float gfr = vbuf[r][512 + d];
        const float gu  = vbuf[r][768 + d];
        const float go  = vbuf[r][1024 + d];
        const float cl = cpool[ctrl[2 + r]][d];
        const float cr = cpool[ctrl[4 + r]][d];
        const float c2 = cl * sigf(gfl + 1.0f) + cr * sigf(gfr + 1.0f) + tanhf(gu) * sigf(gi);
        sh2h[r][d] = sigf(go) * tanhf(c2);
        sh2c[r][d] = c2;
      }
    }
    __syncthreads();

    // (e) reference's edge-only .at[].add refresh
    {
      const int d = tid;
      if (ctrl[7]) { hpool[ctrl[9]][d] += sh2h[1][d]; cpool[ctrl[9]][d] += sh2c[1][d]; }
      if (ctrl[6]) { hpool[ctrl[8]][d] += sh2h[0][d]; cpool[ctrl[8]][d] += sh2c[0][d]; }
    }
    __syncthreads();
  }

  // ---------- final step (i = 46) + outputs
  const int frow = (47 < mylen) ? posComp[0] : posH[0];
  out[(size_t)b * 256 + tid] = hpool[frow][tid];
  out[8192 + (size_t)b * 256 + tid] = cpool[frow][tid];
  if (tid < 46) out[16384 + tid * 32 + b] = (float)allIdx[tid];
  if (tid == 0) {
    out[16384 + 1472 + b] = scal[0];
    out[16384 + 1504 + b] = scal[1] / (float)mylen;
  }
}

extern "C" void kernel_launch(void* const* d_in, const int* in_sizes, int n_in,
                              void* d_out, int out_size, void* d_ws, size_t ws_size,
                              hipStream_t stream) {
  (void)in_sizes; (void)n_in; (void)out_size; (void)d_ws; (void)ws_size;
  const float* x      = (const float*)d_in[0];
  const int*   length = (const int*)d_in[1];
  const float* Ww     = (const float*)d_in[2];
  const float* bw     = (const float*)d_in[3];
  const float* Wc     = (const float*)d_in[4];
  const float* bc     = (const float*)d_in[5];
  const float* q      = (const float*)d_in[6];
  float* out = (float*)d_out;
  coop_latent_tree_kernel<<<dim3(NB), dim3(256), 0, stream>>>(x, length, Ww, bw, Wc, bc, q, out);
}